// pRotatE_3264175145000
// MI455X (gfx1250) — compile-verified
//
#include <hip/hip_runtime.h>
#include <math.h>

#define PI_F 3.14159265358979323846f

typedef __attribute__((ext_vector_type(2))) float v2f;
typedef __attribute__((ext_vector_type(8))) float v8f;

static constexpr int EMB   = 256;
static constexpr int BATCH = 64;
static constexpr int NENT  = 50000;

// ws layout:
//   bytes [0..7]   : uint max_ent_bits, uint max_rel_bits
//   bytes [256..)  : A_cos  (64*256 f32)  =  cos(phase_head+phase_rel)
//   next 64KB      : A_nsin (64*256 f32)  = -sin(phase_head+phase_rel)

__global__ void pr_init_kernel(unsigned int* ws) {
    if (threadIdx.x == 0) { ws[0] = 0u; ws[1] = 0u; }
}

__global__ void pr_maxabs_kernel(const float* __restrict__ ent,
                                 const float* __restrict__ rel,
                                 int n_ent, int n_rel,
                                 unsigned int* ws) {
    __shared__ float se[256];
    __shared__ float sr[256];
    float me = 0.0f, mr = 0.0f;
    const int stride = gridDim.x * blockDim.x;
    for (int i = blockIdx.x * blockDim.x + threadIdx.x; i < n_ent; i += stride)
        me = fmaxf(me, fabsf(ent[i]));
    for (int i = blockIdx.x * blockDim.x + threadIdx.x; i < n_rel; i += stride)
        mr = fmaxf(mr, fabsf(rel[i]));
    se[threadIdx.x] = me; sr[threadIdx.x] = mr;
    __syncthreads();
    for (int s = blockDim.x >> 1; s > 0; s >>= 1) {
        if ((int)threadIdx.x < s) {
            se[threadIdx.x] = fmaxf(se[threadIdx.x], se[threadIdx.x + s]);
            sr[threadIdx.x] = fmaxf(sr[threadIdx.x], sr[threadIdx.x + s]);
        }
        __syncthreads();
    }
    if (threadIdx.x == 0) {
        atomicMax(&ws[0], __float_as_uint(se[0]));
        atomicMax(&ws[1], __float_as_uint(sr[0]));
    }
}

__global__ void pr_prepA_kernel(const int* __restrict__ triples,
                                const float* __restrict__ ent,
                                const float* __restrict__ rel,
                                const unsigned int* __restrict__ ws,
                                float* __restrict__ Ac,
                                float* __restrict__ Ans) {
    const int idx = blockIdx.x * blockDim.x + threadIdx.x;   // 64*256 threads
    const int b = idx >> 8;
    const int k = idx & 255;
    const float inv_e = PI_F / __uint_as_float(ws[0]);
    const float inv_r = PI_F / __uint_as_float(ws[1]);
    const int h = triples[b * 3 + 0];
    const int r = triples[b * 3 + 1];
    const float ph = ent[(size_t)h * EMB + k] * inv_e +
                     rel[(size_t)r * EMB + k] * inv_r;
    Ac[idx]  =  __cosf(ph);
    Ans[idx] = -__sinf(ph);
}

// Each wave: 16 entities (N) x all 64 batch rows (4 M-tiles of 16).
// out[b,e] = sum_k cos(ph[b,k])*sin(pe[e,k]) + (-sin(ph[b,k]))*cos(pe[e,k])
// computed with V_WMMA_F32_16X16X4_F32, full fp32 precision.
__global__ void __launch_bounds__(256)
pr_score_kernel(const float* __restrict__ ent,
                const unsigned int* __restrict__ ws,
                const float* __restrict__ Ac,
                const float* __restrict__ Ans,
                float* __restrict__ out) {
    const int lane   = threadIdx.x & 31;
    const int wave   = threadIdx.x >> 5;
    const int n_base = blockIdx.x * 128 + wave * 16;
    if (n_base >= NENT) return;                 // wave-uniform: EXEC stays all-ones

    const float inv_e = PI_F / __uint_as_float(ws[0]);
    const int l15   = lane & 15;
    const int khalf = (lane >> 4) * 2;          // lanes 0-15: K+{0,1}; 16-31: K+{2,3}

    const float* __restrict__ brow = ent + (size_t)(n_base + l15) * EMB + khalf;

    v8f acc[4];
    #pragma unroll
    for (int mt = 0; mt < 4; ++mt) acc[mt] = (v8f){0.f,0.f,0.f,0.f,0.f,0.f,0.f,0.f};

    for (int k0 = 0; k0 < EMB; k0 += 4) {
        const float p0 = brow[k0 + 0] * inv_e;
        const float p1 = brow[k0 + 1] * inv_e;
        v2f bs; bs.x = __sinf(p0); bs.y = __sinf(p1);
        v2f bc; bc.x = __cosf(p0); bc.y = __cosf(p1);
        #pragma unroll
        for (int mt = 0; mt < 4; ++mt) {
            const size_t aoff = (size_t)(mt * 16 + l15) * EMB + khalf + k0;
            v2f a_c;  a_c.x  = Ac[aoff];  a_c.y  = Ac[aoff + 1];
            v2f a_ns; a_ns.x = Ans[aoff]; a_ns.y = Ans[aoff + 1];
            acc[mt] = __builtin_amdgcn_wmma_f32_16x16x4_f32(
                false, a_c, false, bs, (short)0, acc[mt], false, false);
            acc[mt] = __builtin_amdgcn_wmma_f32_16x16x4_f32(
                false, a_ns, false, bc, (short)0, acc[mt], false, false);
        }
    }

    // D layout: VGPR j, lanes 0-15 -> M = mt*16 + j ; lanes 16-31 -> M = mt*16 + 8 + j
    const int ncol = n_base + l15;
    #pragma unroll
    for (int mt = 0; mt < 4; ++mt) {
        const int mbase = mt * 16 + (lane >> 4) * 8;
        #pragma unroll
        for (int j = 0; j < 8; ++j)
            out[(size_t)(mbase + j) * NENT + ncol] = acc[mt][j];
    }
}

extern "C" void kernel_launch(void* const* d_in, const int* in_sizes, int n_in,
                              void* d_out, int out_size, void* d_ws, size_t ws_size,
                              hipStream_t stream) {
    const int*   triples = (const int*)d_in[0];
    const float* ent     = (const float*)d_in[1];
    const float* rel     = (const float*)d_in[2];
    float* out = (float*)d_out;

    unsigned int* wsm = (unsigned int*)d_ws;
    float* Ac  = (float*)((char*)d_ws + 256);
    float* Ans = (float*)((char*)d_ws + 256 + (size_t)BATCH * EMB * sizeof(float));

    const int n_ent = in_sizes[1];
    const int n_rel = in_sizes[2];

    hipLaunchKernelGGL(pr_init_kernel, dim3(1), dim3(32), 0, stream, wsm);
    hipLaunchKernelGGL(pr_maxabs_kernel, dim3(1024), dim3(256), 0, stream,
                       ent, rel, n_ent, n_rel, wsm);
    hipLaunchKernelGGL(pr_prepA_kernel, dim3((BATCH * EMB) / 256), dim3(256), 0, stream,
                       triples, ent, rel, wsm, Ac, Ans);
    hipLaunchKernelGGL(pr_score_kernel, dim3((NENT + 127) / 128), dim3(256), 0, stream,
                       ent, wsm, Ac, Ans, out);
}